// MambaBlock2D_56324201120367
// MI455X (gfx1250) — compile-verified
//
#include <hip/hip_runtime.h>
#include <math.h>

// ---- problem constants -----------------------------------------------------
#define BATCH   8
#define DIM     96
#define LSEQ    4096          // 64*64
#define EINNER  192           // D_INNER
#define NSTATE  16
#define DTRANK  6
#define NPROJ   384           // 2*D_INNER

typedef __attribute__((ext_vector_type(2))) float v2f;
typedef __attribute__((ext_vector_type(8))) float v8f;

// ---------------------------------------------------------------------------
// K1: in_proj  —  xz[b,l,:] = u[b,l,:] @ W^T,  u[b,l,c] = x[b,c,l]
// GEMM M=B*L=32768, N=384, K=96 via V_WMMA_F32_16X16X4_F32.
// One wave computes a 16x16 tile; 4 waves/block cover 4 N-tiles.
// ---------------------------------------------------------------------------
__global__ __launch_bounds__(128) void k_in_proj(const float* __restrict__ x,
                                                 const float* __restrict__ W,
                                                 float* __restrict__ xm,
                                                 float* __restrict__ z) {
    const int lane = threadIdx.x & 31;
    const int wave = threadIdx.x >> 5;
    const int half = lane >> 4;       // 0: K pair (k0,k0+1)  1: (k0+2,k0+3)
    const int idx  = lane & 15;       // M index for A, N index for B/D

    const int mtile = blockIdx.x;          // 2048 tiles, 256 per batch image
    const int b     = mtile >> 8;
    const int l0    = (mtile & 255) << 4;
    const int n0    = ((blockIdx.y << 2) + wave) << 4;   // 0..368

    const float* xb = x + (size_t)b * DIM * LSEQ;

    v8f acc = {};
    for (int k0 = 0; k0 < DIM; k0 += 4) {
        const int ka = k0 + 2 * half;
        v2f a, bb;
        a.x  = xb[(size_t)ka       * LSEQ + l0 + idx];   // coalesced along L
        a.y  = xb[(size_t)(ka + 1) * LSEQ + l0 + idx];
        bb.x = W[(n0 + idx) * DIM + ka];
        bb.y = W[(n0 + idx) * DIM + ka + 1];
        acc  = __builtin_amdgcn_wmma_f32_16x16x4_f32(false, a, false, bb,
                                                     (short)0, acc, false, false);
    }

    const int n = n0 + idx;
#pragma unroll
    for (int r = 0; r < 8; ++r) {
        const int    m   = r + 8 * half;
        const size_t row = (size_t)b * LSEQ + l0 + m;
        const float  v   = acc[r];
        if (n < EINNER) xm[row * EINNER + n] = v;
        else            z [row * EINNER + (n - EINNER)] = v;
    }
}

// ---------------------------------------------------------------------------
// K2: depthwise causal conv (width 4) along L + SiLU.  (B,L,E) layout.
// ---------------------------------------------------------------------------
__global__ __launch_bounds__(256) void k_conv_silu(const float* __restrict__ xm,
                                                   const float* __restrict__ cw,
                                                   const float* __restrict__ cb,
                                                   float* __restrict__ xa) {
    const size_t i = (size_t)blockIdx.x * 256 + threadIdx.x;
    if (i >= (size_t)BATCH * LSEQ * EINNER) return;
    const int    e    = (int)(i % EINNER);
    const size_t bl   = i / EINNER;
    const int    l    = (int)(bl & (LSEQ - 1));
    const size_t base = (bl - l) * EINNER + e;   // (b,0,e)

    float acc = cb[e];
#pragma unroll
    for (int k = 0; k < 4; ++k) {
        const int lt = l - 3 + k;
        if (lt >= 0) acc += cw[e * 4 + k] * xm[base + (size_t)lt * EINNER];
    }
    xa[i] = acc * (1.0f / (1.0f + expf(-acc)));   // SiLU
}

// ---------------------------------------------------------------------------
// K3: x_proj  —  dbl[b,l,j] = xa[b,l,:] . x_proj_w[j,:],  j<38. One block per
// (b,l): row staged in LDS, 38 active lanes do the K=192 dots.
// ---------------------------------------------------------------------------
__global__ __launch_bounds__(64) void k_x_proj(const float* __restrict__ xa,
                                               const float* __restrict__ W,
                                               float* __restrict__ dt,
                                               float* __restrict__ Bm,
                                               float* __restrict__ Cm) {
    __shared__ float row[EINNER];
    const size_t bl  = blockIdx.x;
    const int    tid = threadIdx.x;
    for (int i = tid; i < EINNER; i += 64) row[i] = xa[bl * EINNER + i];
    __syncthreads();
    if (tid < DTRANK + 2 * NSTATE) {
        const float* w = W + tid * EINNER;
        float acc = 0.0f;
        for (int e = 0; e < EINNER; ++e) acc += row[e] * w[e];
        if (tid < DTRANK)           dt[bl * DTRANK + tid]                = acc;
        else if (tid < DTRANK + NSTATE) Bm[bl * NSTATE + (tid - DTRANK)] = acc;
        else                        Cm[bl * NSTATE + (tid - DTRANK - NSTATE)] = acc;
    }
}

// ---------------------------------------------------------------------------
// K4: delta = softplus(dt @ dt_proj_w^T + dt_proj_b)
// ---------------------------------------------------------------------------
__global__ __launch_bounds__(256) void k_dt_proj(const float* __restrict__ dt,
                                                 const float* __restrict__ W,
                                                 const float* __restrict__ bias,
                                                 float* __restrict__ delta) {
    const size_t i = (size_t)blockIdx.x * 256 + threadIdx.x;
    if (i >= (size_t)BATCH * LSEQ * EINNER) return;
    const int    e  = (int)(i % EINNER);
    const size_t bl = i / EINNER;
    float acc = bias[e];
#pragma unroll
    for (int r = 0; r < DTRANK; ++r) acc += dt[bl * DTRANK + r] * W[e * DTRANK + r];
    delta[i] = (acc > 20.0f) ? acc : log1pf(expf(acc));
}

// ---------------------------------------------------------------------------
// K5: selective scan (serial over L) fused with D-skip and SiLU(z) gating.
// One block per batch, 192 threads (6 wave32) = one state row h[16] each.
// B_t / C_t shared across channels via double-buffered LDS (1 barrier/step).
// ---------------------------------------------------------------------------
__global__ __launch_bounds__(EINNER) void k_scan(const float* __restrict__ delta,
                                                 const float* __restrict__ u,
                                                 const float* __restrict__ zb,
                                                 const float* __restrict__ Bm,
                                                 const float* __restrict__ Cm,
                                                 const float* __restrict__ A_log,
                                                 const float* __restrict__ Dp,
                                                 float* __restrict__ y) {
    __shared__ float Bs[2][NSTATE];
    __shared__ float Cs[2][NSTATE];

    const int    e   = threadIdx.x;              // 0..191
    const int    b   = blockIdx.x;
    const size_t bl0 = (size_t)b * LSEQ;

    float Arow[NSTATE];
#pragma unroll
    for (int n = 0; n < NSTATE; ++n) Arow[n] = -expf(A_log[e * NSTATE + n]);
    const float dskip = Dp[e];

    float h[NSTATE];
#pragma unroll
    for (int n = 0; n < NSTATE; ++n) h[n] = 0.0f;

    const float* Bbase = Bm + bl0 * NSTATE;
    const float* Cbase = Cm + bl0 * NSTATE;

    // preload t = 0
    if (e < NSTATE)          Bs[0][e]          = Bbase[e];
    else if (e < 2 * NSTATE) Cs[0][e - NSTATE] = Cbase[e - NSTATE];
    __syncthreads();

    for (int t = 0; t < LSEQ; ++t) {
        const int cur = t & 1, nxt = cur ^ 1;
        if (t + 1 < LSEQ) {
            if (e < NSTATE)          Bs[nxt][e]          = Bbase[(size_t)(t + 1) * NSTATE + e];
            else if (e < 2 * NSTATE) Cs[nxt][e - NSTATE] = Cbase[(size_t)(t + 1) * NSTATE + (e - NSTATE)];
        }
        const size_t ix = (bl0 + t) * EINNER + e;
        if (t + 8 < LSEQ) __builtin_prefetch(&delta[ix + 8 * EINNER], 0, 1);

        const float dt_ = delta[ix];
        const float u_  = u[ix];
        const float zv  = zb[ix];

        float acc = 0.0f;
        const float du = dt_ * u_;
#pragma unroll
        for (int n = 0; n < NSTATE; ++n) {
            const float dA = expf(dt_ * Arow[n]);
            h[n] = dA * h[n] + du * Bs[cur][n];
            acc += h[n] * Cs[cur][n];
        }
        acc += u_ * dskip;
        const float gate = zv * (1.0f / (1.0f + expf(-zv)));   // SiLU(z)
        y[ix] = acc * gate;
        __syncthreads();   // protects reuse of buffer `cur` next iteration
    }
}

// ---------------------------------------------------------------------------
// K6: out_proj  —  GEMM M=32768, N=96, K=192 via V_WMMA_F32_16X16X4_F32.
// ---------------------------------------------------------------------------
__global__ __launch_bounds__(128) void k_out_proj(const float* __restrict__ y,
                                                  const float* __restrict__ W,
                                                  float* __restrict__ outp) {
    const int lane  = threadIdx.x & 31;
    const int wave  = threadIdx.x >> 5;
    const int half  = lane >> 4;
    const int idx   = lane & 15;
    const int ntile = (blockIdx.y << 2) + wave;   // grid.y=2 -> 0..7
    if (ntile >= DIM / 16) return;                // wave-uniform guard (EXEC all-1)

    const size_t m0 = (size_t)blockIdx.x * 16;
    const int    n0 = ntile << 4;

    v8f acc = {};
    for (int k0 = 0; k0 < EINNER; k0 += 4) {
        const int ka = k0 + 2 * half;
        v2f a, bb;
        a.x  = y[(m0 + idx) * EINNER + ka];
        a.y  = y[(m0 + idx) * EINNER + ka + 1];
        bb.x = W[(n0 + idx) * EINNER + ka];
        bb.y = W[(n0 + idx) * EINNER + ka + 1];
        acc  = __builtin_amdgcn_wmma_f32_16x16x4_f32(false, a, false, bb,
                                                     (short)0, acc, false, false);
    }
#pragma unroll
    for (int r = 0; r < 8; ++r) {
        const size_t m = m0 + r + 8 * half;
        outp[m * DIM + n0 + idx] = acc[r];
    }
}

// ---------------------------------------------------------------------------
// K7: LayerNorm over DIM + transpose (B,L,C) -> (B,C,H,W). Thread per (b,l).
// ---------------------------------------------------------------------------
__global__ __launch_bounds__(256) void k_layernorm(const float* __restrict__ outp,
                                                   const float* __restrict__ gamma,
                                                   const float* __restrict__ beta,
                                                   float* __restrict__ out) {
    const size_t bl = (size_t)blockIdx.x * 256 + threadIdx.x;
    if (bl >= (size_t)BATCH * LSEQ) return;
    const int b = (int)(bl >> 12);
    const int l = (int)(bl & (LSEQ - 1));
    const float* r = outp + bl * DIM;

    float mu = 0.0f;
    for (int c = 0; c < DIM; ++c) mu += r[c];
    mu *= (1.0f / DIM);
    float var = 0.0f;
    for (int c = 0; c < DIM; ++c) { const float d = r[c] - mu; var += d * d; }
    var *= (1.0f / DIM);
    const float inv = rsqrtf(var + 1e-5f);

    for (int c = 0; c < DIM; ++c)
        out[((size_t)b * DIM + c) * LSEQ + l] = (r[c] - mu) * inv * gamma[c] + beta[c];
}

// ---------------------------------------------------------------------------
extern "C" void kernel_launch(void* const* d_in, const int* in_sizes, int n_in,
                              void* d_out, int out_size, void* d_ws, size_t ws_size,
                              hipStream_t stream) {
    const float* x         = (const float*)d_in[0];
    const float* in_proj_w = (const float*)d_in[1];
    const float* conv_w    = (const float*)d_in[2];
    const float* conv_b    = (const float*)d_in[3];
    const float* x_proj_w  = (const float*)d_in[4];
    const float* dt_proj_w = (const float*)d_in[5];
    const float* dt_proj_b = (const float*)d_in[6];
    const float* A_log     = (const float*)d_in[7];
    const float* Dvec      = (const float*)d_in[8];
    const float* out_proj_w= (const float*)d_in[9];
    const float* ln_gamma  = (const float*)d_in[10];
    const float* ln_beta   = (const float*)d_in[11];

    float* ws = (float*)d_ws;
    const size_t NBLE = (size_t)BATCH * LSEQ * EINNER;      // 6291456
    float* xm_raw = ws;                                     // reused as y after scan
    float* zbuf   = ws + NBLE;                              // reused as out_pre
    float* xm_act = ws + 2 * NBLE;
    float* deltab = ws + 3 * NBLE;
    float* dtbuf  = ws + 4 * NBLE;                          // B*L*6
    float* Bmbuf  = dtbuf + (size_t)BATCH * LSEQ * DTRANK;  // B*L*16
    float* Cmbuf  = Bmbuf + (size_t)BATCH * LSEQ * NSTATE;  // B*L*16
    float* ybuf   = xm_raw;   // xm_raw dead after conv
    float* outpre = zbuf;     // z dead after scan gating

    const int MT = (BATCH * LSEQ) / 16;   // 2048 M-tiles

    // 1) in_proj (WMMA f32)
    k_in_proj<<<dim3(MT, NPROJ / 64), 128, 0, stream>>>(x, in_proj_w, xm_raw, zbuf);
    // 2) conv + SiLU
    k_conv_silu<<<(unsigned)((NBLE + 255) / 256), 256, 0, stream>>>(xm_raw, conv_w, conv_b, xm_act);
    // 3) x_proj
    k_x_proj<<<BATCH * LSEQ, 64, 0, stream>>>(xm_act, x_proj_w, dtbuf, Bmbuf, Cmbuf);
    // 4) dt_proj + softplus
    k_dt_proj<<<(unsigned)((NBLE + 255) / 256), 256, 0, stream>>>(dtbuf, dt_proj_w, dt_proj_b, deltab);
    // 5) selective scan + gating
    k_scan<<<BATCH, EINNER, 0, stream>>>(deltab, xm_act, zbuf, Bmbuf, Cmbuf, A_log, Dvec, ybuf);
    // 6) out_proj (WMMA f32)
    k_out_proj<<<dim3(MT, 2), 128, 0, stream>>>(ybuf, out_proj_w, outpre);
    // 7) LayerNorm + NCHW transpose
    k_layernorm<<<(BATCH * LSEQ + 255) / 256, 256, 0, stream>>>(outpre, ln_gamma, ln_beta, (float*)d_out);
}